// XSimGCL_51874615001253
// MI455X (gfx1250) — compile-verified
//
#include <hip/hip_runtime.h>

// Problem constants (match reference)
#define NP      300000          // NUM_PERSONS
#define NH      60000           // NUM_HOBBIES
#define NN      (NP + NH)       // total nodes
#define DIM     64
#define NLAYERS 3
#define NNZ_C   2000000
#define BATCH   16384

typedef __attribute__((ext_vector_type(2))) float v2f;
typedef __attribute__((ext_vector_type(8))) float v8f;

// Native CDNA5 L2 float atomic (no-return form -> STOREcnt; S_ENDPGM waits idle).
__device__ __forceinline__ void gatomic_add_f32(float* p, float v) {
  asm volatile("global_atomic_add_f32 %0, %1, off" : : "v"(p), "v"(v) : "memory");
}

// cur = concat(person_emb, hobby_emb), float4-vectorized.
__global__ void k_concat(const float4* __restrict__ pe4, const float4* __restrict__ he4,
                         float4* __restrict__ cur4, int total4) {
  int i = blockIdx.x * blockDim.x + threadIdx.x;
  if (i >= total4) return;
  const int pf = NP * (DIM / 4);
  cur4[i] = (i < pf) ? pe4[i] : he4[i - pf];
}

// Layer-0 contribution: pAcc[b] = person_emb[pid[b]], hAcc[b] = hobby_emb[hid[b]].
__global__ void k_init_acc(const int* __restrict__ pid, const int* __restrict__ hid,
                           const float4* __restrict__ pe4, const float4* __restrict__ he4,
                           float4* __restrict__ pAcc, float4* __restrict__ hAcc) {
  int t = blockIdx.x * blockDim.x + threadIdx.x;
  const int per = BATCH * (DIM / 4);
  if (t >= 2 * per) return;
  int part = (t >= per);
  int s = t - part * per;
  int b = s >> 4;           // DIM/4 == 16 float4 per row
  int c = s & 15;
  if (!part) pAcc[s] = pe4[(size_t)pid[b] * 16 + c];
  else       hAcc[s] = he4[(size_t)hid[b] * 16 + c];
}

// SpMM scatter: nxt[row] += val * cur[col]. 16 lanes per edge, float4 per lane
// (coalesced 256B row access, 4 native f32 atomics per lane). cur/nxt live in L2.
__global__ void k_scatter(const int* __restrict__ rows, const int* __restrict__ cols,
                          const float* __restrict__ vals, const float* __restrict__ cur,
                          float* __restrict__ nxt, int nnz) {
  unsigned long long t = (unsigned long long)blockIdx.x * blockDim.x + threadIdx.x;
  unsigned e = (unsigned)(t >> 4);
  unsigned c = ((unsigned)t & 15u) << 2;
  if (e >= (unsigned)nnz) return;
  int   r   = rows[e];
  int   col = cols[e];
  float v   = vals[e];
  const float4 x = *reinterpret_cast<const float4*>(cur + (size_t)col * DIM + c);
  float* dst = nxt + (size_t)r * DIM + c;
  gatomic_add_f32(dst + 0, v * x.x);
  gatomic_add_f32(dst + 1, v * x.y);
  gatomic_add_f32(dst + 2, v * x.z);
  gatomic_add_f32(dst + 3, v * x.w);
}

// After each layer: pAcc[b] += nxt[pid[b]], hAcc[b] += nxt[NP + hid[b]].
__global__ void k_gather_acc(const int* __restrict__ pid, const int* __restrict__ hid,
                             const float4* __restrict__ nxt4,
                             float4* __restrict__ pAcc, float4* __restrict__ hAcc) {
  int t = blockIdx.x * blockDim.x + threadIdx.x;
  const int per = BATCH * (DIM / 4);
  if (t >= 2 * per) return;
  int part = (t >= per);
  int s = t - part * per;
  int b = s >> 4;
  int c = s & 15;
  size_t row = part ? ((size_t)NP + (size_t)hid[b]) : (size_t)pid[b];
  float4 add = nxt4[row * 16 + c];
  float4* dst = part ? (hAcc + s) : (pAcc + s);
  float4 o = *dst;
  o.x += add.x; o.y += add.y; o.z += add.z; o.w += add.w;
  *dst = o;
}

// Final dot products via V_WMMA_F32_16X16X4_F32 (exact fp32 path).
// One wave handles 16 outputs: D = (P/4)(16x64) x (H/4)^T(64x16) chained over 16
// K=4 steps; extract the 16x16 diagonal. B divisible by 16 -> EXEC all-1s always.
__global__ void k_dot_wmma(const float* __restrict__ pAcc, const float* __restrict__ hAcc,
                           float* __restrict__ out) {
  int wave = blockIdx.x * (blockDim.x >> 5) + (threadIdx.x >> 5);
  int lane = threadIdx.x & 31;
  int m    = lane & 15;              // A row (M) / B column (N) held by this lane
  int koff = (lane >> 4) << 1;       // lanes 0-15: K {0,1}; lanes 16-31: K {2,3}

  const float* prow = pAcc + ((size_t)wave * 16 + m) * DIM;
  const float* hrow = hAcc + ((size_t)wave * 16 + m) * DIM;

  v8f c = {};
#pragma unroll
  for (int kc = 0; kc < 16; ++kc) {
    int k = (kc << 2) + koff;
    v2f a, b;
    a.x = prow[k]     * 0.25f;       // fold avg = acc/4 into each operand
    a.y = prow[k + 1] * 0.25f;
    b.x = hrow[k]     * 0.25f;
    b.y = hrow[k + 1] * 0.25f;
    c = __builtin_amdgcn_wmma_f32_16x16x4_f32(
        /*neg_a=*/false, a, /*neg_b=*/false, b,
        /*c_mod=*/(short)0, c, /*reuse_a=*/false, /*reuse_b=*/false);
  }

  // Diagonal extraction per 16x16 f32 C/D layout:
  // lane group 0 (0-15) holds M = r, lane group 1 (16-31) holds M = r + 8.
  int n = lane & 15;
  int r = n - ((lane >> 4) << 3);
  if (r >= 0 && r < 8) {
    out[(size_t)wave * 16 + n] = c[r];
  }
}

extern "C" void kernel_launch(void* const* d_in, const int* in_sizes, int n_in,
                              void* d_out, int out_size, void* d_ws, size_t ws_size,
                              hipStream_t stream) {
  (void)in_sizes; (void)n_in; (void)out_size; (void)ws_size;

  const int*   person_ids = (const int*)  d_in[0];
  const int*   hobby_ids  = (const int*)  d_in[1];
  const int*   adj_rows   = (const int*)  d_in[2];
  const int*   adj_cols   = (const int*)  d_in[3];
  const float* adj_vals   = (const float*)d_in[4];
  const float* person_emb = (const float*)d_in[5];
  const float* hobby_emb  = (const float*)d_in[6];
  float*       out        = (float*)d_out;

  // Workspace layout (all 256B aligned): cur | nxt | pAcc | hAcc  (~192.6 MB)
  char* ws = (char*)d_ws;
  const size_t nodeBytes = (size_t)NN * DIM * sizeof(float);   // 92.16 MB
  float* bufA = (float*)ws;
  float* bufB = (float*)(ws + nodeBytes);
  float* pAcc = (float*)(ws + 2 * nodeBytes);
  float* hAcc = pAcc + (size_t)BATCH * DIM;

  // 1) cur = concat(embeddings); pAcc/hAcc = layer-0 sampled rows.
  {
    int total4 = NN * (DIM / 4);
    k_concat<<<(total4 + 255) / 256, 256, 0, stream>>>(
        (const float4*)person_emb, (const float4*)hobby_emb, (float4*)bufA, total4);
    int tot = 2 * BATCH * (DIM / 4);
    k_init_acc<<<(tot + 255) / 256, 256, 0, stream>>>(
        person_ids, hobby_ids, (const float4*)person_emb, (const float4*)hobby_emb,
        (float4*)pAcc, (float4*)hAcc);
  }

  // 2) Three propagation layers: zero, scatter-add, sampled accumulate, swap.
  float* cur = bufA;
  float* nxt = bufB;
  for (int l = 0; l < NLAYERS; ++l) {
    hipMemsetAsync(nxt, 0, nodeBytes, stream);
    unsigned long long tot = (unsigned long long)NNZ_C * 16ull;
    k_scatter<<<(unsigned)((tot + 255) / 256), 256, 0, stream>>>(
        adj_rows, adj_cols, adj_vals, cur, nxt, NNZ_C);
    int tg = 2 * BATCH * (DIM / 4);
    k_gather_acc<<<(tg + 255) / 256, 256, 0, stream>>>(
        person_ids, hobby_ids, (const float4*)nxt, (float4*)pAcc, (float4*)hAcc);
    float* tmp = cur; cur = nxt; nxt = tmp;
  }

  // 3) WMMA batched dot: 1024 tiles of 16 pairs, 8 waves per 256-thread block.
  k_dot_wmma<<<(BATCH / 16) / 8, 256, 0, stream>>>(pAcc, hAcc, out);
}